// Set2Set_51754355917417
// MI455X (gfx1250) — compile-verified
//
#include <hip/hip_runtime.h>
#include <math.h>

// Problem constants (match reference)
constexpr int kH   = 128;        // IN_FEATS
constexpr int kB   = 1024;       // NUM_GRAPHS
constexpr int kK   = 3 * kH;     // GEMM K dim: [Sx | cnt*h | h] = 384
constexpr int kG   = 4 * kH;     // gates width = 512
constexpr int kSplit = 8;        // partial-sum split per graph
constexpr int kIters = 3;

typedef __attribute__((ext_vector_type(2))) float v2f;
typedef __attribute__((ext_vector_type(8))) float v8f;

__device__ __forceinline__ float sigm(float v) { return 1.0f / (1.0f + expf(-v)); }

__device__ __forceinline__ int lower_bound_i32(const int* __restrict__ a, int n, int val) {
    int lo = 0, hi = n;
    while (lo < hi) {
        int mid = (lo + hi) >> 1;
        if (a[mid] < val) lo = mid + 1; else hi = mid;
    }
    return lo;
}

// ---- init: h = c = 0 ------------------------------------------------------
__global__ void zero_hc(float* __restrict__ h, float* __restrict__ c) {
    int i = blockIdx.x * blockDim.x + threadIdx.x;
    if (i < kB * kH) { h[i] = 0.0f; c[i] = 0.0f; }
}

// ---- pack W = [W_ih | W_hh] row-major (kG x kK) ---------------------------
__global__ void pack_w(const float* __restrict__ W_ih, const float* __restrict__ W_hh,
                       float* __restrict__ Wc) {
    int i = blockIdx.x * blockDim.x + threadIdx.x;
    if (i >= kG * kK) return;
    int j = i / kK, k = i - j * kK;
    Wc[i] = (k < 2 * kH) ? W_ih[j * (2 * kH) + k] : W_hh[j * kH + (k - 2 * kH)];
}

// ---- segment sum of x over sorted batch -----------------------------------
// 8-way split per graph; block = 4 waves; each lane owns a float4 feature
// quad (global_load_b128 -> 512B per wave per row), each wave strides rows.
// Per-wave partials combined through LDS in fixed order (deterministic).
__global__ void seg_partial(const float* __restrict__ x, const int* __restrict__ batch,
                            int N, float* __restrict__ Sxp, float* __restrict__ cnt) {
    __shared__ float4 red[4][32];
    int b = blockIdx.x >> 3;           // graph id
    int s = blockIdx.x & (kSplit - 1); // split id
    int start = lower_bound_i32(batch, N, b);
    int end   = lower_bound_i32(batch, N, b + 1);
    int len   = end - start;
    int chunk = (len + kSplit - 1) / kSplit;
    int r0 = start + s * chunk;
    int r1 = min(end, r0 + chunk);
    int lane = threadIdx.x & 31;       // feature quad: floats [4*lane, 4*lane+4)
    int wv   = threadIdx.x >> 5;       // wave id: row phase

    float4 a0 = make_float4(0.f, 0.f, 0.f, 0.f);
    float4 a1 = make_float4(0.f, 0.f, 0.f, 0.f);
    int r = r0 + wv;
    for (; r + 4 < r1; r += 8) {
        float4 v0 = *(const float4*)(x + (size_t)r * kH + lane * 4);
        float4 v1 = *(const float4*)(x + (size_t)(r + 4) * kH + lane * 4);
        a0.x += v0.x; a0.y += v0.y; a0.z += v0.z; a0.w += v0.w;
        a1.x += v1.x; a1.y += v1.y; a1.z += v1.z; a1.w += v1.w;
    }
    for (; r < r1; r += 4) {
        float4 v0 = *(const float4*)(x + (size_t)r * kH + lane * 4);
        a0.x += v0.x; a0.y += v0.y; a0.z += v0.z; a0.w += v0.w;
    }
    a0.x += a1.x; a0.y += a1.y; a0.z += a1.z; a0.w += a1.w;

    red[wv][lane] = a0;
    __syncthreads();
    if (wv == 0) {
        float4 p0 = red[0][lane], p1 = red[1][lane], p2 = red[2][lane], p3 = red[3][lane];
        float4 sum = make_float4((p0.x + p1.x) + (p2.x + p3.x),
                                 (p0.y + p1.y) + (p2.y + p3.y),
                                 (p0.z + p1.z) + (p2.z + p3.z),
                                 (p0.w + p1.w) + (p2.w + p3.w));
        *(float4*)(Sxp + ((size_t)s * kB + b) * kH + lane * 4) = sum;
        if (lane == 0 && s == 0) cnt[b] = (float)len;
    }
}

// ---- fold 8 partials -> Sx, float4-vectorized, fixed order ----------------
__global__ void seg_reduce(const float* __restrict__ Sxp, float* __restrict__ Sx) {
    int q = blockIdx.x * blockDim.x + threadIdx.x;   // quad index
    if (q >= kB * kH / 4) return;
    float4 s = make_float4(0.f, 0.f, 0.f, 0.f);
#pragma unroll
    for (int p = 0; p < kSplit; ++p) {
        float4 v = *(const float4*)(Sxp + (size_t)p * kB * kH + (size_t)q * 4);
        s.x += v.x; s.y += v.y; s.z += v.z; s.w += v.w;
    }
    *(float4*)(Sx + (size_t)q * 4) = s;
}

// ---- build A = [Sx | cnt*h | h]  (kB x kK) --------------------------------
__global__ void build_A(const float* __restrict__ Sx, const float* __restrict__ cnt,
                        const float* __restrict__ h, float* __restrict__ A) {
    int i = blockIdx.x * blockDim.x + threadIdx.x;
    if (i >= kB * kK) return;
    int b = i / kK, k = i - b * kK;
    float v;
    if (k < kH)          v = Sx[b * kH + k];
    else if (k < 2 * kH) v = cnt[b] * h[b * kH + (k - kH)];
    else                 v = h[b * kH + (k - 2 * kH)];
    A[i] = v;
}

// ---- gates = A @ Wc^T + b_ih + b_hh via V_WMMA_F32_16X16X4_F32 ------------
// One wave per 16x16 output tile. 64 m-tiles x 32 n-tiles = 2048 waves.
// A fragment (16x4 f32, ISA 7.12.2): lane = 16*(k/2) + m ; VGPR v -> k = 2*(lane/16)+v
// B fragment (4x16 f32, symmetric): lane = 16*(k/2) + n ; VGPR v -> k = 2*(lane/16)+v
// C/D (16x16 f32): lane -> N = lane&15 ; VGPR v -> M = v + 8*(lane>>4)
__global__ void gemm_wmma(const float* __restrict__ A, const float* __restrict__ Wc,
                          const float* __restrict__ b_ih, const float* __restrict__ b_hh,
                          float* __restrict__ gates) {
    int w    = blockIdx.x * (blockDim.x >> 5) + (threadIdx.x >> 5);
    int lane = threadIdx.x & 31;
    int mt = w >> 5;   // 0..63  (rows of B=1024)
    int nt = w & 31;   // 0..31  (cols of gates=512)
    int mrow  = mt * 16 + (lane & 15);
    int ncol  = nt * 16 + (lane & 15);
    int khalf = (lane >> 4) * 2;
    const float* Ap = A  + (size_t)mrow * kK + khalf;
    const float* Bp = Wc + (size_t)ncol * kK + khalf;
    v8f acc = {};
#pragma unroll 4
    for (int k0 = 0; k0 < kK; k0 += 4) {
        v2f afrag = *(const v2f*)(Ap + k0);
        v2f bfrag = *(const v2f*)(Bp + k0);
        acc = __builtin_amdgcn_wmma_f32_16x16x4_f32(
            /*neg_a=*/false, afrag, /*neg_b=*/false, bfrag,
            /*c_mod=*/(short)0, acc, /*reuse_a=*/false, /*reuse_b=*/false);
    }
    float bias = b_ih[ncol] + b_hh[ncol];
    int rbase = mt * 16 + (lane >> 4) * 8;
#pragma unroll
    for (int v = 0; v < 8; ++v)
        gates[(size_t)(rbase + v) * kG + ncol] = acc[v] + bias;
}

// ---- LSTM cell elementwise; also writes out = [h, h] on last iteration ----
__global__ void lstm_update(const float* __restrict__ gates, float* __restrict__ h,
                            float* __restrict__ c, float* __restrict__ out, int write_out) {
    int i = blockIdx.x * blockDim.x + threadIdx.x;
    if (i >= kB * kH) return;
    int b = i >> 7, t = i & (kH - 1);
    const float* g = gates + (size_t)b * kG;
    float ig = sigm(g[t]);
    float fg = sigm(g[kH + t]);
    float gg = tanhf(g[2 * kH + t]);
    float og = sigm(g[3 * kH + t]);
    float cv = fg * c[i] + ig * gg;
    float hv = og * tanhf(cv);
    c[i] = cv;
    h[i] = hv;
    if (write_out) {
        out[(size_t)b * (2 * kH) + t]      = hv;  // q_star
        out[(size_t)b * (2 * kH) + kH + t] = hv;  // h (== q_star)
    }
}

extern "C" void kernel_launch(void* const* d_in, const int* in_sizes, int n_in,
                              void* d_out, int out_size, void* d_ws, size_t ws_size,
                              hipStream_t stream) {
    const float* x     = (const float*)d_in[0];
    const int*   batch = (const int*)  d_in[1];
    const float* W_ih  = (const float*)d_in[2];
    const float* W_hh  = (const float*)d_in[3];
    const float* b_ih  = (const float*)d_in[4];
    const float* b_hh  = (const float*)d_in[5];
    float* out = (float*)d_out;
    int N = in_sizes[0] / kH;

    // Workspace layout (floats); all offsets multiples of 4 -> float4 aligned.
    float* ws  = (float*)d_ws;
    float* Sxp = ws;                                    // kSplit*kB*kH
    float* Sx  = Sxp + (size_t)kSplit * kB * kH;        // kB*kH
    float* cnt = Sx  + (size_t)kB * kH;                 // kB
    float* h   = cnt + kB;                              // kB*kH
    float* c   = h   + (size_t)kB * kH;                 // kB*kH
    float* A   = c   + (size_t)kB * kH;                 // kB*kK
    float* Wc  = A   + (size_t)kB * kK;                 // kG*kK
    float* gts = Wc  + (size_t)kG * kK;                 // kB*kG

    zero_hc<<<(kB * kH + 255) / 256, 256, 0, stream>>>(h, c);
    pack_w<<<(kG * kK + 255) / 256, 256, 0, stream>>>(W_ih, W_hh, Wc);
    seg_partial<<<kB * kSplit, 128, 0, stream>>>(x, batch, N, Sxp, cnt);
    seg_reduce<<<(kB * kH / 4 + 255) / 256, 256, 0, stream>>>(Sxp, Sx);

    for (int it = 0; it < kIters; ++it) {
        build_A<<<(kB * kK + 255) / 256, 256, 0, stream>>>(Sx, cnt, h, A);
        // 2048 tile-waves, 4 waves (128 threads) per block -> 512 blocks
        gemm_wmma<<<(kB / 16) * (kG / 16) / 4, 128, 0, stream>>>(A, Wc, b_ih, b_hh, gts);
        lstm_update<<<(kB * kH + 255) / 256, 256, 0, stream>>>(gts, h, c, out,
                                                               it == kIters - 1 ? 1 : 0);
    }
}